// MGDPR_33363305955514
// MI455X (gfx1250) — compile-verified
//
#include <hip/hip_runtime.h>

typedef __attribute__((ext_vector_type(2))) float v2f;
typedef __attribute__((ext_vector_type(8))) float v8f;

#define GAMMA_INV 0.4f   // 1/2.5, strictly-lower-triangular retention decay

// ---------------------------------------------------------------------------
// Generic fp32 WMMA GEMM:  C = epi( A @ B (+bias) )
//   A: [M,K] row-major (lda), optionally elementwise-multiplied by A2 (AMUL)
//   B: TRANS_B ? [Ncol,K] row-major (ldb)  :  [K,Ncol] row-major (ldb)
//   C: [M,Ncol] row-major (ldc)
// grid = (M/64, Ncol/64, batch); block = (32,4). Each wave: 16 rows x 64 cols
// (one A fragment feeds 4 v_wmma_f32_16x16x4_f32 per K-step).
// EPI: 0 none | 1 +bias | 2 +bias,PReLU(alpha from device) | 3 D_gamma mask
// ---------------------------------------------------------------------------
template<int EPI, bool TRANS_B, bool AMUL>
__global__ __launch_bounds__(128) void wmma_gemm(
    const float* __restrict__ A,  long long sA,
    const float* __restrict__ A2, long long sA2,
    const float* __restrict__ B,  long long sB,
    const float* __restrict__ bias, long long sBias,
    float* __restrict__ C, long long sC,
    int K, int lda, int ldb, int ldc,
    const float* __restrict__ alphaPtr)
{
  const int bb = blockIdx.z;
  const float* Ab  = A + (long long)bb * sA;
  const float* A2b = AMUL ? (A2 + (long long)bb * sA2) : nullptr;
  const float* Bb  = B + (long long)bb * sB;
  float* Cb = C + (long long)bb * sC;

  const int lane = threadIdx.x;      // 0..31 (wave32)
  const int wave = threadIdx.y;      // 0..3
  const int half = lane >> 4;        // K-half selector for A/B fragments
  const int l16  = lane & 15;
  const int rowBase = (blockIdx.x * 4 + wave) * 16;
  const int colBase = blockIdx.y * 64;

  v8f acc[4];
#pragma unroll
  for (int j = 0; j < 4; ++j)
#pragma unroll
    for (int v = 0; v < 8; ++v) acc[j][v] = 0.f;

  const int arow = rowBase + l16;
  for (int k0 = 0; k0 < K; k0 += 4) {
    const int kk = k0 + 2 * half;
    // A fragment (16x4): lane half h holds K = {2h, 2h+1} of its row
    const float* ap = Ab + (long long)arow * lda + kk;
    v2f a;
    a.x = ap[0];
    a.y = ap[1];
    if (AMUL) {  // fused S * a_batch for the diffusion GEMM
      const float* mp = A2b + (long long)arow * lda + kk;
      a.x *= mp[0];
      a.y *= mp[1];
    }
#pragma unroll
    for (int j = 0; j < 4; ++j) {
      const int col = colBase + j * 16 + l16;
      v2f b;
      if (TRANS_B) {
        const float* bp = Bb + (long long)col * ldb + kk;
        b.x = bp[0];
        b.y = bp[1];
      } else {
        b.x = Bb[(long long)kk * ldb + col];
        b.y = Bb[(long long)(kk + 1) * ldb + col];
      }
      acc[j] = __builtin_amdgcn_wmma_f32_16x16x4_f32(
          false, a, false, b, (short)0, acc[j], false, false);
    }
  }

  float alpha = 0.f;
  if (EPI == 2) alpha = alphaPtr[0];
  const float* biasb = nullptr;
  if (EPI == 1 || EPI == 2) biasb = bias + (long long)bb * sBias;

#pragma unroll
  for (int j = 0; j < 4; ++j) {
    const int col = colBase + j * 16 + l16;
    float bv = (EPI == 1 || EPI == 2) ? biasb[col] : 0.f;
#pragma unroll
    for (int v = 0; v < 8; ++v) {
      const int row = rowBase + v + 8 * half;   // C/D layout: VGPR v -> M = v + 8*half
      float val = acc[j][v];
      if (EPI == 1 || EPI == 2) val += bv;
      if (EPI == 2) val = (val >= 0.f) ? val : alpha * val;
      if (EPI == 3) val = (row > col) ? val * GAMMA_INV : 0.f;
      Cb[(long long)row * ldc + col] = val;
    }
  }
}

// S[r,n,m] = sum_e theta[r,e] * T[r,e,n,m]   (R=5, E=3, N=1024)
__global__ void compute_S_kernel(const float* __restrict__ T,
                                 const float* __restrict__ theta,
                                 float* __restrict__ S)
{
  const long long NN = 1024LL * 1024LL;
  long long i = (long long)blockIdx.x * blockDim.x + threadIdx.x;   // < 5*NN
  int r = (int)(i / NN);
  long long off = i - (long long)r * NN;
  const float* Tp = T + (long long)r * 3 * NN + off;
  S[i] = theta[r * 3 + 0] * Tp[0]
       + theta[r * 3 + 1] * Tp[NN]
       + theta[r * 3 + 2] * Tp[2 * NN];
}

// u[b,s,n,o] = prelu( sum_r cw[s,r]*diff[b,r,n,o] + cb[s], a1 )
__global__ void conv_mix_kernel(const float* __restrict__ diff,
                                const float* __restrict__ cw,
                                const float* __restrict__ cb,
                                const float* __restrict__ a1p,
                                float* __restrict__ u)
{
  int i = blockIdx.x * blockDim.x + threadIdx.x;    // < B*R*N*64 = 655360
  int b = i / 327680;
  int rem = i - b * 327680;
  int s = rem >> 16;          // relation (N*64 = 65536)
  int no = rem & 65535;       // n*64 + o
  float acc = cb[s];
  const float* dp = diff + b * 327680 + no;
#pragma unroll
  for (int r = 0; r < 5; ++r) acc += cw[s * 5 + r] * dp[r << 16];
  float a1 = a1p[0];
  u[i] = (acc >= 0.f) ? acc : a1 * acc;
}

// skip0[b,n,r*64+f] = x[b,r,n,f]
__global__ void skip0_kernel(const float* __restrict__ x, float* __restrict__ skip)
{
  int i = blockIdx.x * blockDim.x + threadIdx.x;    // < B*N*320 = 655360
  int b = i / 327680;
  int rem = i - b * 327680;
  int n = rem / 320;
  int c = rem - n * 320;
  int r = c >> 6;
  int f = c & 63;
  skip[i] = x[((b * 5 + r) * 1024 + n) * 64 + f];
}

// out[b,n,c] = hid[b,n,:] @ mlp2_w[c,:] + mlp2_b[c]   (N_CLS=2, H1=128)
__global__ void mlp_out_kernel(const float* __restrict__ hid,
                               const float* __restrict__ w2,
                               const float* __restrict__ b2,
                               float* __restrict__ out)
{
  int i = blockIdx.x * blockDim.x + threadIdx.x;    // < B*N = 2048
  const float* hp = hid + i * 128;
  float o0 = b2[0], o1 = b2[1];
#pragma unroll 8
  for (int j = 0; j < 128; ++j) {
    float hv = hp[j];
    o0 += w2[j] * hv;
    o1 += w2[128 + j] * hv;
  }
  out[i * 2 + 0] = o0;
  out[i * 2 + 1] = o1;
}

extern "C" void kernel_launch(void* const* d_in, const int* in_sizes, int n_in,
                              void* d_out, int out_size, void* d_ws, size_t ws_size,
                              hipStream_t stream)
{
  (void)in_sizes; (void)n_in; (void)out_size; (void)ws_size;
  const float* x_batch = (const float*)d_in[0];
  const float* a_batch = (const float*)d_in[1];
  const float* T       = (const float*)d_in[2];
  const float* theta   = (const float*)d_in[3];
  const float* fc_w    = (const float*)d_in[4];
  const float* fc_b    = (const float*)d_in[5];
  const float* conv_w  = (const float*)d_in[6];
  const float* conv_b  = (const float*)d_in[7];
  const float* a0      = (const float*)d_in[8];
  const float* a1      = (const float*)d_in[9];
  const float* qw      = (const float*)d_in[10];
  const float* qb      = (const float*)d_in[11];
  const float* kw      = (const float*)d_in[12];
  const float* kb      = (const float*)d_in[13];
  const float* vw      = (const float*)d_in[14];
  const float* vb      = (const float*)d_in[15];
  const float* rw      = (const float*)d_in[16];
  const float* rb      = (const float*)d_in[17];
  const float* a_ret   = (const float*)d_in[18];
  const float* l1_w    = (const float*)d_in[19];
  const float* l1_b    = (const float*)d_in[20];
  const float* l2_w    = (const float*)d_in[21];
  const float* l2_b    = (const float*)d_in[22];
  const float* mlp1_w  = (const float*)d_in[23];
  const float* mlp1_b  = (const float*)d_in[24];
  const float* mlp2_w  = (const float*)d_in[25];
  const float* mlp2_b  = (const float*)d_in[26];
  const float* a_mlp   = (const float*)d_in[27];
  float* out = (float*)d_out;

  // workspace layout (floats); ~55 MB total
  float* ws = (float*)d_ws;
  float* S     = ws;  ws += 5242880;   // [R,N,N]
  float* feats = ws;  ws += 655360;    // [B,R,N,64]
  float* diffb = ws;  ws += 655360;    // [B,R,N,64]
  float* ub    = ws;  ws += 655360;    // [B,R,N,64] (== xr view [B,N,320])
  float* qx    = ws;  ws += 524288;    // [B,N,256]
  float* kx    = ws;  ws += 524288;
  float* vx    = ws;  ws += 524288;
  float* sc    = ws;  ws += 2097152;   // [B,N,N] (D_gamma-masked scores)
  float* rt    = ws;  ws += 524288;    // [B,N,256]
  float* cat   = ws;  ws += 786432;    // [B,N,384] = [eta | ts]
  float* skipA = ws;  ws += 655360;    // [B,N,320]
  float* skipB = ws;  ws += 655360;
  float* hid   = ws;  ws += 262144;    // [B,N,128]

  const long long NN = 1024LL * 1024LL;
  dim3 blk(32, 4);

  skip0_kernel<<<2560, 256, 0, stream>>>(x_batch, skipA);

  float* skip_cur = skipA;
  float* skip_nxt = skipB;

  for (int l = 0; l < 2; ++l) {
    const float* h_in = (l == 0) ? x_batch : ub;
    compute_S_kernel<<<20480, 256, 0, stream>>>(T + (long long)l * 5 * 3 * NN,
                                                theta + l * 15, S);
    for (int b = 0; b < 2; ++b) {
      // feats[b,r] = (S[r] .* a[b,r]) @ h[b,r]   (M=1024,N=64,K=1024, batch r)
      wmma_gemm<0, false, true><<<dim3(16, 1, 5), blk, 0, stream>>>(
          S, NN,
          a_batch + (long long)b * 5 * NN, NN,
          h_in + (long long)b * 5 * 1024 * 64, 1024 * 64,
          nullptr, 0,
          feats + (long long)b * 5 * 1024 * 64, 1024 * 64,
          1024, 1024, 64, 64, nullptr);
      // diff = prelu(feats @ fc_w^T + fc_b, a0)   (M=1024,N=64,K=64, batch r)
      wmma_gemm<2, true, false><<<dim3(16, 1, 5), blk, 0, stream>>>(
          feats + (long long)b * 5 * 1024 * 64, 1024 * 64, nullptr, 0,
          fc_w + (long long)l * 5 * 64 * 64, 64 * 64,
          fc_b + l * 5 * 64, 64,
          diffb + (long long)b * 5 * 1024 * 64, 1024 * 64,
          64, 64, 64, 64, a0 + l);
    }
    conv_mix_kernel<<<2560, 256, 0, stream>>>(diffb, conv_w + l * 25,
                                              conv_b + l * 5, a1 + l, ub);
    // q/k/v = xr @ W^T + b   (xr == ub viewed [2048,320]; M=2048,N=256,K=320)
    wmma_gemm<1, true, false><<<dim3(32, 4, 1), blk, 0, stream>>>(
        ub, 0, nullptr, 0, qw + (long long)l * 256 * 320, 0, qb + l * 256, 0,
        qx, 0, 320, 320, 320, 256, nullptr);
    wmma_gemm<1, true, false><<<dim3(32, 4, 1), blk, 0, stream>>>(
        ub, 0, nullptr, 0, kw + (long long)l * 256 * 320, 0, kb + l * 256, 0,
        kx, 0, 320, 320, 320, 256, nullptr);
    wmma_gemm<1, true, false><<<dim3(32, 4, 1), blk, 0, stream>>>(
        ub, 0, nullptr, 0, vw + (long long)l * 256 * 320, 0, vb + l * 256, 0,
        vx, 0, 320, 320, 320, 256, nullptr);
    // sc = D_gamma .* (q @ k^T)   (M=1024,N=1024,K=256, batch b)
    wmma_gemm<3, true, false><<<dim3(16, 16, 2), blk, 0, stream>>>(
        qx, 1024 * 256, nullptr, 0, kx, 1024 * 256, nullptr, 0,
        sc, NN, 256, 256, 256, 1024, nullptr);
    // rt = sc @ v   (M=1024,N=256,K=1024, batch b)
    wmma_gemm<0, false, false><<<dim3(16, 4, 2), blk, 0, stream>>>(
        sc, NN, nullptr, 0, vx, 1024 * 256, nullptr, 0,
        rt, 1024 * 256, 1024, 1024, 256, 256, nullptr);
    // eta = prelu(rt @ rw^T + rb, a_ret) -> cat[:, 0:256]
    wmma_gemm<2, true, false><<<dim3(32, 4, 1), blk, 0, stream>>>(
        rt, 0, nullptr, 0, rw + (long long)l * 256 * 256, 0, rb + l * 256, 0,
        cat, 0, 256, 256, 256, 384, a_ret + l);
    // ts = skip @ l1_w^T + l1_b -> cat[:, 256:384]
    wmma_gemm<1, true, false><<<dim3(32, 2, 1), blk, 0, stream>>>(
        skip_cur, 0, nullptr, 0, l1_w + (long long)l * 128 * 320, 0, l1_b + l * 128, 0,
        cat + 256, 0, 320, 320, 320, 384, nullptr);
    // skip' = cat @ l2_w^T + l2_b   (M=2048,N=320,K=384)
    wmma_gemm<1, true, false><<<dim3(32, 5, 1), blk, 0, stream>>>(
        cat, 0, nullptr, 0, l2_w + (long long)l * 320 * 384, 0, l2_b + l * 320, 0,
        skip_nxt, 0, 384, 384, 384, 320, nullptr);
    float* t = skip_cur; skip_cur = skip_nxt; skip_nxt = t;
  }

  // hid = prelu(skip @ mlp1_w^T + mlp1_b, a_mlp)   (M=2048,N=128,K=320)
  wmma_gemm<2, true, false><<<dim3(32, 2, 1), blk, 0, stream>>>(
      skip_cur, 0, nullptr, 0, mlp1_w, 0, mlp1_b, 0,
      hid, 0, 320, 320, 320, 128, a_mlp);
  mlp_out_kernel<<<8, 256, 0, stream>>>(hid, mlp2_w, mlp2_b, out);
}